// SPDNet_80693845557395
// MI455X (gfx1250) — compile-verified
//
#include <hip/hip_runtime.h>

// SPDNet forward for MI455X (gfx1250, wave32, WMMA f32 16x16x4 + TDM).
//   X[512,128,1000] --(P = W^T X, Gram of centered P)--> Y[512,64,64]
//   Y --(parallel Jacobi eig, log-clamp recompose, linear head)--> out[512,4]

typedef float v2f __attribute__((ext_vector_type(2)));
typedef float v8f __attribute__((ext_vector_type(8)));
typedef unsigned int u32x4 __attribute__((ext_vector_type(4)));
typedef unsigned int u32x8 __attribute__((ext_vector_type(8)));

#define B_N 512
#define C_N 128
#define T_N 1000
#define D_N 64
#define TC 64                 // time-chunk per LDS stage
#define NCHUNK 16             // ceil(1000/64), tail zero-filled by TDM OOB
#define XC_STRIDE (TC + 2)    // 66  (Xc[c][t], padded by TDM: 2 dwords / row)
#define PS_STRIDE (TC + 2)    // 66  (Ps[d][t])
#define VS_STRIDE (D_N + 2)   // 66  (As/Vs/Vw [row][col])
#define NSWEEP 12

// dynamic LDS: double-buffered X tile + P tile
#define SMEM_COV_FLOATS (2 * C_N * XC_STRIDE + D_N * PS_STRIDE)

__device__ __forceinline__ v8f wmma_f32(v2f a, v2f b, v8f c) {
  // D = A(16x4) * B(4x16) + C(16x16), f32, wave32
  return __builtin_amdgcn_wmma_f32_16x16x4_f32(false, a, false, b, (short)0, c,
                                               false, false);
}

// TDM: DMA a [128 x 64] f32 tile (row stride T_N) into LDS with 2-dword row
// padding (LDS row stride 66) and OOB zero-fill past rem_t columns.
// Must be executed by exactly one wave (EXEC is ignored by TENSOR ops).
__device__ __forceinline__ void tdm_load_tile(const float* gsrc,
                                              unsigned lds_off, int rem_t) {
  const unsigned long long ga = (unsigned long long)(uintptr_t)gsrc;
  u32x4 g0;
  g0[0] = 1u;                                   // count=1 (valid user D#)
  g0[1] = lds_off;                              // lds_addr (bytes)
  g0[2] = (unsigned)ga;                         // global_addr[31:0]
  g0[3] = ((unsigned)(ga >> 32) & 0x01FFFFFFu)  // global_addr[56:32]
          | (2u << 30);                         // type=2 (image)
  u32x8 g1;
  g1[0] = (2u << 16)       // data_size = 4 bytes
          | (1u << 20)     // pad_enable
          | (5u << 22)     // pad_interval: 64 dwords
          | (1u << 25);    // pad_amount: 2 dwords -> row stride 66
  g1[1] = ((unsigned)rem_t & 0xFFFFu) << 16;    // tensor_dim0 (OOB -> 0)
  g1[2] = ((unsigned)C_N) << 16;                // tensor_dim1 = 128
  g1[3] = ((unsigned)TC) << 16;                 // tile_dim0 = 64
  g1[4] = (unsigned)C_N;                        // tile_dim1 = 128
  g1[5] = (unsigned)T_N;                        // tensor_dim0_stride = 1000
  g1[6] = 0u;
  g1[7] = 0u;
  u32x4 g2 = {};  // 2D tile: groups 2/3 unused
  u32x4 g3 = {};
  asm volatile("tensor_load_to_lds %0, %1, %2, %3"
               :
               : "s"(g0), "s"(g1), "s"(g2), "s"(g3)
               : "memory");
}

// ---------------------------------------------------------------------------
// Kernel 0: transpose W [C,D] -> Wt [D,C] so A-fragments read k-contiguously.
// ---------------------------------------------------------------------------
__global__ __launch_bounds__(256) void spdnet_wt_kernel(
    const float* __restrict__ W, float* __restrict__ Wt) {
  int idx = blockIdx.x * 256 + threadIdx.x;
  if (idx < C_N * D_N) {
    int c = idx >> 6, d = idx & 63;
    Wt[d * C_N + c] = W[idx];
  }
}

// ---------------------------------------------------------------------------
// Kernel 1: one workgroup (8 waves) per batch sample.
//   Double-buffered TDM staging of X chunks [128c x 64t] overlapped with
//   compute; P = Wt*X via WMMA (Wt fragments held in VGPRs); Gram of P
//   accumulated via WMMA; Y = Gram/T - mu*mu^T.
// ---------------------------------------------------------------------------
__global__ __launch_bounds__(256) void spdnet_cov_kernel(
    const float* __restrict__ X, const float* __restrict__ Wt,
    float* __restrict__ Y) {
  extern __shared__ float smem[];
  float* const XcBuf0 = smem;                          // 128*66
  float* const XcBuf1 = smem + C_N * XC_STRIDE;        // 128*66
  float* const Psm    = smem + 2 * C_N * XC_STRIDE;    // 64*66
  __shared__ float red[256];
  __shared__ float psums[D_N];

  const int b = blockIdx.x;
  const int tid = threadIdx.x;
  const int wave = tid >> 5;
  const int lane = tid & 31;
  const int l16 = lane & 15;
  const int khalf = (lane >> 4) << 1;  // k sub-offset 0 / 2 per half-wave

  const int mi = wave & 3;        // output row tile (d), also Gram gi
  const int nA = wave >> 2;       // this wave owns col tiles {nA, nA+2}
  const int m0 = mi * 16;
  const int tn0 = nA * 16;
  const int tn1 = (nA + 2) * 16;
  const int mrow = m0 + ((lane >> 4) << 3);

  // ---- preload all Wt A-fragments for this wave's row tile (K = 128) ----
  v2f awt[C_N / 4];
#pragma unroll
  for (int kk4 = 0; kk4 < C_N / 4; ++kk4)
    awt[kk4] = *(const v2f*)(Wt + (m0 + l16) * C_N + kk4 * 4 + khalf);

  v8f gacc0 = {};
  v8f gacc1 = {};
  float psum_part = 0.0f;
  const int sd = tid & 63;   // d index for partial row sums
  const int sq = tid >> 6;   // time-quarter 0..3

  const float* Xb = X + (size_t)b * C_N * T_N;

  // prime the pipeline: DMA chunk 0 into buffer 0
  if (wave == 0) tdm_load_tile(Xb, (unsigned)(uintptr_t)XcBuf0, T_N);

  for (int ch = 0; ch < NCHUNK; ++ch) {
    // ---- wait for this chunk's DMA; it overlapped previous compute ----
    if (wave == 0) __builtin_amdgcn_s_wait_tensorcnt(0);
    __syncthreads();
    // ---- kick off next chunk's DMA into the other buffer ----
    if (wave == 0 && ch + 1 < NCHUNK) {
      const int t1 = (ch + 1) * TC;
      float* nxt = ((ch + 1) & 1) ? XcBuf1 : XcBuf0;
      tdm_load_tile(Xb + t1, (unsigned)(uintptr_t)nxt, T_N - t1);
    }
    const float* Xc = (ch & 1) ? XcBuf1 : XcBuf0;

    // ---- projection: P tile (mi, {tn0,tn1}) = Wt * Xchunk, K = 128 ----
    v8f pacc0 = {};
    v8f pacc1 = {};
#pragma unroll
    for (int kk = 0; kk < C_N; kk += 4) {
      const float* xb = Xc + (kk + khalf) * XC_STRIDE;
      v2f b0, b1;
      b0[0] = xb[tn0 + l16];
      b0[1] = xb[XC_STRIDE + tn0 + l16];
      b1[0] = xb[tn1 + l16];
      b1[1] = xb[XC_STRIDE + tn1 + l16];
      pacc0 = wmma_f32(awt[kk >> 2], b0, pacc0);
      pacc1 = wmma_f32(awt[kk >> 2], b1, pacc1);
    }
    // write D fragments -> Ps[d][t]  (row M = r + 8*(lane/16), col N = l16)
#pragma unroll
    for (int r = 0; r < 8; ++r) {
      Psm[(mrow + r) * PS_STRIDE + tn0 + l16] = pacc0[r];
      Psm[(mrow + r) * PS_STRIDE + tn1 + l16] = pacc1[r];
    }
    __syncthreads();

    // ---- Gram accumulation: tiles (mi, nA) and (mi, nA+2), K = 64 t ----
#pragma unroll 4
    for (int kk = 0; kk < TC; kk += 4) {
      v2f a  = *(const v2f*)(&Psm[(m0 + l16) * PS_STRIDE + kk + khalf]);
      v2f b0 = *(const v2f*)(&Psm[(tn0 + l16) * PS_STRIDE + kk + khalf]);
      v2f b1 = *(const v2f*)(&Psm[(tn1 + l16) * PS_STRIDE + kk + khalf]);
      gacc0 = wmma_f32(a, b0, gacc0);
      gacc1 = wmma_f32(a, b1, gacc1);
    }
    // ---- partial row sums of P (for mean correction) ----
#pragma unroll
    for (int j = 0; j < 16; ++j)
      psum_part += Psm[sd * PS_STRIDE + sq * 16 + j];
    // next iteration's top barrier protects Ps reuse; the buffer written by
    // the in-flight DMA was last read one chunk ago (two barriers back).
  }

  // reduce row sums
  red[tid] = psum_part;
  __syncthreads();
  if (tid < 64)
    psums[tid] = red[tid] + red[tid + 64] + red[tid + 128] + red[tid + 192];
  __syncthreads();

  // Y = Gram/T - mu*mu^T
  const float invT = 1.0f / (float)T_N;
  float* Yb = Y + ((size_t)b << 12);
#pragma unroll
  for (int r = 0; r < 8; ++r) {
    const int i = mrow + r;
    const float mu_i = psums[i] * invT;
    const int j0 = tn0 + l16, j1 = tn1 + l16;
    Yb[i * 64 + j0] = gacc0[r] * invT - mu_i * (psums[j0] * invT);
    Yb[i * 64 + j1] = gacc1[r] * invT - mu_i * (psums[j1] * invT);
  }
}

// ---------------------------------------------------------------------------
// Kernel 2: one workgroup per sample.
//   Parallel cyclic Jacobi eig of 64x64 symmetric Y (round-robin tournament,
//   32 disjoint rotations/round), fused ReEig+LogEig recompose via WMMA,
//   fused 4-way linear head.
// ---------------------------------------------------------------------------
__device__ __forceinline__ int rr_player(int pos, int r) {
  return (pos == 0) ? 0 : 1 + (pos - 1 + r) % 63;
}

__global__ __launch_bounds__(256) void spdnet_eig_kernel(
    const float* __restrict__ Yin, const float* __restrict__ lin_w,
    const float* __restrict__ lin_b, float* __restrict__ out) {
  __shared__ __attribute__((aligned(16))) float As[D_N * VS_STRIDE];
  __shared__ __attribute__((aligned(16))) float Vs[D_N * VS_STRIDE];
  __shared__ __attribute__((aligned(16))) float Vw[D_N * VS_STRIDE];
  __shared__ float csC[32], csS[32];
  __shared__ float lw[D_N];
  __shared__ float red4[8][4];

  const int b = blockIdx.x;
  const int tid = threadIdx.x;
  const float* Yb = Yin + ((size_t)b << 12);

  // load A, V = I
  for (int e = tid; e < 4096; e += 256) {
    int i = e >> 6, j = e & 63;
    As[i * VS_STRIDE + j] = Yb[e];
    Vs[i * VS_STRIDE + j] = (i == j) ? 1.0f : 0.0f;
  }
  __syncthreads();

  const int pair = tid & 31;
  const int k0 = (tid >> 5) * 8;  // each thread updates 8 row/col elements

  for (int sw = 0; sw < NSWEEP; ++sw) {
    for (int r = 0; r < 63; ++r) {
      if (tid < 32) {
        int p = rr_player(tid, r);
        int q = rr_player(63 - tid, r);
        float app = As[p * VS_STRIDE + p];
        float aqq = As[q * VS_STRIDE + q];
        float apq = As[p * VS_STRIDE + q];
        float c = 1.0f, s = 0.0f;
        if (fabsf(apq) > 1e-12f) {
          float tau = (aqq - app) / (2.0f * apq);
          float t = copysignf(1.0f, tau) /
                    (fabsf(tau) + sqrtf(1.0f + tau * tau));
          c = 1.0f / sqrtf(1.0f + t * t);
          s = t * c;
        }
        csC[tid] = c;
        csS[tid] = s;
      }
      __syncthreads();
      const int p = rr_player(pair, r);
      const int q = rr_player(63 - pair, r);
      const float c = csC[pair];
      const float s = csS[pair];
      // phase 1: rows p,q of A (disjoint across pairs)
#pragma unroll
      for (int jj = 0; jj < 8; ++jj) {
        int k = k0 + jj;
        float ap = As[p * VS_STRIDE + k];
        float aq = As[q * VS_STRIDE + k];
        As[p * VS_STRIDE + k] = c * ap - s * aq;
        As[q * VS_STRIDE + k] = s * ap + c * aq;
      }
      __syncthreads();
      // phase 2: cols p,q of A and V (disjoint across pairs)
#pragma unroll
      for (int jj = 0; jj < 8; ++jj) {
        int k = k0 + jj;
        float ap = As[k * VS_STRIDE + p];
        float aq = As[k * VS_STRIDE + q];
        As[k * VS_STRIDE + p] = c * ap - s * aq;
        As[k * VS_STRIDE + q] = s * ap + c * aq;
        float vp = Vs[k * VS_STRIDE + p];
        float vq = Vs[k * VS_STRIDE + q];
        Vs[k * VS_STRIDE + p] = c * vp - s * vq;
        Vs[k * VS_STRIDE + q] = s * vp + c * vq;
      }
      __syncthreads();
    }
  }

  // fused ReEig + LogEig spectrum: lw = log(max(lambda, 1e-4))
  if (tid < 64) lw[tid] = logf(fmaxf(As[tid * VS_STRIDE + tid], 1e-4f));
  __syncthreads();
  // Vw = V * diag(lw)
  for (int e = tid; e < 4096; e += 256) {
    int i = e >> 6, k = e & 63;
    Vw[i * VS_STRIDE + k] = Vs[i * VS_STRIDE + k] * lw[k];
  }
  __syncthreads();

  // Ylog = Vw * V^T via WMMA (16 tiles, 2 per wave), result into As
  {
    const int wave = tid >> 5, lane = tid & 31;
    const int l16 = lane & 15, kh = (lane >> 4) << 1;
    const int ti = wave & 3;
    const int tj0 = (wave >> 2) * 16;
    const int tj1 = tj0 + 32;
    v8f acc0 = {};
    v8f acc1 = {};
#pragma unroll 4
    for (int kk = 0; kk < 64; kk += 4) {
      v2f a  = *(const v2f*)(&Vw[(ti * 16 + l16) * VS_STRIDE + kk + kh]);
      v2f b0 = *(const v2f*)(&Vs[(tj0 + l16) * VS_STRIDE + kk + kh]);
      v2f b1 = *(const v2f*)(&Vs[(tj1 + l16) * VS_STRIDE + kk + kh]);
      acc0 = wmma_f32(a, b0, acc0);
      acc1 = wmma_f32(a, b1, acc1);
    }
    const int irow = ti * 16 + ((lane >> 4) << 3);
#pragma unroll
    for (int r = 0; r < 8; ++r) {
      As[(irow + r) * VS_STRIDE + tj0 + l16] = acc0[r];
      As[(irow + r) * VS_STRIDE + tj1 + l16] = acc1[r];
    }
  }
  __syncthreads();

  // linear head: out[b][o] = sum_f Ylog[f] * lin_w[o][f] + lin_b[o]
  float acc[4] = {0.f, 0.f, 0.f, 0.f};
  for (int e = tid; e < 4096; e += 256) {
    int i = e >> 6, j = e & 63;
    float v = As[i * VS_STRIDE + j];
#pragma unroll
    for (int o = 0; o < 4; ++o) acc[o] += v * lin_w[o * 4096 + e];
  }
#pragma unroll
  for (int o = 0; o < 4; ++o)
    for (int off = 16; off > 0; off >>= 1)
      acc[o] += __shfl_down(acc[o], off, 32);
  if ((tid & 31) == 0) {
    int w = tid >> 5;
#pragma unroll
    for (int o = 0; o < 4; ++o) red4[w][o] = acc[o];
  }
  __syncthreads();
  if (tid < 4) {
    float s = lin_b[tid];
#pragma unroll
    for (int w = 0; w < 8; ++w) s += red4[w][tid];
    out[b * 4 + tid] = s;
  }
}

// ---------------------------------------------------------------------------
extern "C" void kernel_launch(void* const* d_in, const int* in_sizes, int n_in,
                              void* d_out, int out_size, void* d_ws,
                              size_t ws_size, hipStream_t stream) {
  const float* X     = (const float*)d_in[0];  // [512,128,1000]
  const float* W     = (const float*)d_in[1];  // [1,128,64]
  const float* lin_w = (const float*)d_in[2];  // [4,4096]
  const float* lin_b = (const float*)d_in[3];  // [4]
  float* out = (float*)d_out;                  // [512,4]

  float* Y  = (float*)d_ws;                    // 512*4096 floats (8.39 MB)
  float* Wt = Y + (size_t)B_N * D_N * D_N;     // 8192 floats

  const size_t smem_cov = (size_t)SMEM_COV_FLOATS * sizeof(float);  // 84480 B
  // best-effort opt-in for >64KB dynamic LDS (no-op / ignored if unsupported)
  (void)hipFuncSetAttribute((const void*)spdnet_cov_kernel,
                            hipFuncAttributeMaxDynamicSharedMemorySize,
                            (int)smem_cov);

  spdnet_wt_kernel<<<(C_N * D_N + 255) / 256, 256, 0, stream>>>(W, Wt);
  spdnet_cov_kernel<<<B_N, 256, smem_cov, stream>>>(X, Wt, Y);
  spdnet_eig_kernel<<<B_N, 256, 0, stream>>>(Y, lin_w, lin_b, out);
}